// TN_60206851555693
// MI455X (gfx1250) — compile-verified
//
#include <hip/hip_runtime.h>
#include <math.h>

typedef float v2f __attribute__((ext_vector_type(2)));
typedef float v8f __attribute__((ext_vector_type(8)));

#define Bq 64
#define Sq 512
#define Rq 64

// ---------------------------------------------------------------------------
// Optional gfx1250 async global->LDS path (ASYNCcnt-tracked DMA-style loads).
// Builtin signature (from compiler diagnostic): 
//   void __builtin_amdgcn_global_load_async_to_lds_b128(
//        int4-vec addrspace(1)* gsrc, int4-vec addrspace(3)* ldst,
//        imm int offset, imm int cpol)
// ---------------------------------------------------------------------------
#if __has_builtin(__builtin_amdgcn_global_load_async_to_lds_b128) && \
    __has_builtin(__builtin_amdgcn_s_wait_asynccnt)
#define HAVE_ASYNC_LDS 1
#endif

#ifdef HAVE_ASYNC_LDS
typedef int v4i_b __attribute__((vector_size(16)));
typedef v4i_b __attribute__((address_space(1)))* g128p;
typedef v4i_b __attribute__((address_space(3)))* l128p;

__device__ __forceinline__ void async_copy_b128(const float* g, float* l) {
    __builtin_amdgcn_global_load_async_to_lds_b128(
        (g128p)g, (l128p)l, /*offset=*/0, /*cpol=*/0);
}
#endif

// ---------------------------------------------------------------------------
// One 16x16 output tile of D = P(64x64) * M(64x64), K=64 as 16 wmma 16x16x4,
// split into two independent accumulator chains for XDL pipelining.
// ---------------------------------------------------------------------------
__device__ __forceinline__ v8f tile_mm(const float* Pl, const float* Ml,
                                       int arow, int bcol, int grp)
{
    v8f acc0 = {}, acc1 = {};
    #pragma unroll
    for (int kb = 0; kb < Rq; kb += 8) {
        v2f a0, b0, a1, b1;
        // A 16x4 layout: VGPR0 = K {0 | 2}, VGPR1 = K {1 | 3} per lane half.
        a0.x = Pl[arow * Rq + kb + 2 * grp + 0];
        a0.y = Pl[arow * Rq + kb + 2 * grp + 1];
        // B 4x16 layout: VGPRj, lane half -> K row j / j+2.
        b0.x = Ml[(kb + 2 * grp + 0) * Rq + bcol];
        b0.y = Ml[(kb + 2 * grp + 1) * Rq + bcol];
        a1.x = Pl[arow * Rq + kb + 4 + 2 * grp + 0];
        a1.y = Pl[arow * Rq + kb + 4 + 2 * grp + 1];
        b1.x = Ml[(kb + 4 + 2 * grp + 0) * Rq + bcol];
        b1.y = Ml[(kb + 4 + 2 * grp + 1) * Rq + bcol];
        acc0 = __builtin_amdgcn_wmma_f32_16x16x4_f32(
                   false, a0, false, b0, (short)0, acc0, false, false);
        acc1 = __builtin_amdgcn_wmma_f32_16x16x4_f32(
                   false, a1, false, b1, (short)0, acc1, false, false);
    }
    return acc0 + acc1;
}

// ---------------------------------------------------------------------------
// Kernel 1: per (segment, batch) compute the left-to-right product of L
// consecutive 64x64 token matrices with V_WMMA_F32_16X16X4_F32.
// Block = 512 threads = 16 wave32 waves; wave (mi,ni) owns output tile (mi,ni).
// ---------------------------------------------------------------------------
__global__ __launch_bounds__(512)
void seg_matchain(const int* __restrict__ ids,
                  const float* __restrict__ G,
                  float* __restrict__ ws,
                  int L, int nseg)
{
    __shared__ float Plds[Rq * Rq];          // running product (16 KB)
#ifdef HAVE_ASYNC_LDS
    __shared__ float Mbuf[2][Rq * Rq];       // double-buffered next matrix
#else
    __shared__ float Mlds[Rq * Rq];
#endif

    const int tid  = threadIdx.x;
    const int lane = tid & 31;
    const int wave = tid >> 5;
    const int mi   = wave >> 2;              // 0..3 row tile
    const int ni   = wave & 3;               // 0..3 col tile
    const int seg  = blockIdx.x;
    const int b    = blockIdx.y;
    const int s0   = seg * L;

    const int half = lane & 15;
    const int grp  = lane >> 4;
    const int arow = mi * 16 + half;
    const int bcol = ni * 16 + half;

    // Load matrix s0 straight into P (coalesced b128 loads, 2 float4/thread).
    {
        int tok = ids[b * Sq + s0];
        const float4* src = (const float4*)(G + (size_t)tok * (Rq * Rq));
        ((float4*)Plds)[tid]       = src[tid];
        ((float4*)Plds)[tid + 512] = src[tid + 512];
    }

#ifdef HAVE_ASYNC_LDS
    // ---- async global->LDS double-buffered pipeline ----
    if (L > 1) {                             // matrix 1 -> Mbuf[1]
        int tok = ids[b * Sq + s0 + 1];
        const float* src = G + (size_t)tok * (Rq * Rq);
        async_copy_b128(src + tid * 4,        &Mbuf[1][tid * 4]);
        async_copy_b128(src + 2048 + tid * 4, &Mbuf[1][2048 + tid * 4]);
    }
    __syncthreads();                         // Plds visible

    for (int s = 1; s < L; ++s) {
        const float* Mcur = Mbuf[s & 1];
        if (s + 1 < L) {                     // kick off matrix s+1 -> other buf
            int tok = ids[b * Sq + s0 + s + 1];
            const float* src = G + (size_t)tok * (Rq * Rq);
            float* dstl = Mbuf[(s + 1) & 1];
            async_copy_b128(src + tid * 4,        dstl + tid * 4);
            async_copy_b128(src + 2048 + tid * 4, dstl + 2048 + tid * 4);
            if (s + 2 < L) {                 // warm L2 two steps ahead
                int tok2 = ids[b * Sq + s0 + s + 2];
                __builtin_prefetch(G + (size_t)tok2 * (Rq * Rq) + tid * 8, 0, 0);
            }
            __builtin_amdgcn_s_wait_asynccnt(2);   // prior matrix landed
        } else {
            __builtin_amdgcn_s_wait_asynccnt(0);
        }
        __syncthreads();                     // everyone's async writes visible

        v8f acc = tile_mm(Plds, Mcur, arow, bcol, grp);
        __syncthreads();                     // all waves done reading P

        #pragma unroll
        for (int j = 0; j < 8; ++j)          // C/D: VGPRj -> rows j / j+8
            Plds[(mi * 16 + j + 8 * grp) * Rq + ni * 16 + half] = acc[j];
    }
    __syncthreads();
#else
    // ---- fallback: register-staged prefetch ----
    float4 r0 = make_float4(0.f, 0.f, 0.f, 0.f), r1 = r0;
    if (L > 1) {
        int tok = ids[b * Sq + s0 + 1];
        const float4* src = (const float4*)(G + (size_t)tok * (Rq * Rq));
        r0 = src[tid];
        r1 = src[tid + 512];
    }
    __syncthreads();

    for (int s = 1; s < L; ++s) {
        ((float4*)Mlds)[tid]       = r0;
        ((float4*)Mlds)[tid + 512] = r1;
        __syncthreads();
        if (s + 1 < L) {
            int tok = ids[b * Sq + s0 + s + 1];
            const float4* src = (const float4*)(G + (size_t)tok * (Rq * Rq));
            r0 = src[tid];
            r1 = src[tid + 512];
            if (s + 2 < L) {
                int tok2 = ids[b * Sq + s0 + s + 2];
                __builtin_prefetch(G + (size_t)tok2 * (Rq * Rq) + tid * 8, 0, 0);
            }
        }
        v8f acc = tile_mm(Plds, Mlds, arow, bcol, grp);
        __syncthreads();
        #pragma unroll
        for (int j = 0; j < 8; ++j)
            Plds[(mi * 16 + j + 8 * grp) * Rq + ni * 16 + half] = acc[j];
        __syncthreads();
    }
#endif

    // Emit the segment product.
    float4* dst = (float4*)(ws + ((size_t)b * nseg + seg) * (Rq * Rq));
    dst[tid]       = ((const float4*)Plds)[tid];
    dst[tid + 512] = ((const float4*)Plds)[tid + 512];
}

// ---------------------------------------------------------------------------
// Kernel 2: per batch, v = G1; v <- v * P_seg for each segment (in order),
// then F.normalize (eps=1e-12), then the 64->2 linear head.
// ---------------------------------------------------------------------------
__global__ __launch_bounds__(64)
void combine_norm(const float* __restrict__ ws,
                  const float* __restrict__ G1,
                  const float* __restrict__ W,
                  const float* __restrict__ bias,
                  float* __restrict__ out,
                  int nseg)
{
    __shared__ float v[Rq];
    __shared__ float red[Rq];
    const int tid = threadIdx.x;
    const int b   = blockIdx.x;

    v[tid] = G1[tid];
    __syncthreads();

    for (int seg = 0; seg < nseg; ++seg) {
        const float* P = ws + ((size_t)b * nseg + seg) * (Rq * Rq);
        float acc = 0.f;
        #pragma unroll 8
        for (int i = 0; i < Rq; ++i)
            acc = fmaf(v[i], P[i * Rq + tid], acc);   // coalesced across tid
        __syncthreads();
        v[tid] = acc;
        __syncthreads();
    }

    const float x = v[tid];
    red[tid] = x * x;
    __syncthreads();
    for (int off = 32; off > 0; off >>= 1) {
        if (tid < off) red[tid] += red[tid + off];
        __syncthreads();
    }
    const float nrm = fmaxf(sqrtf(red[0]), 1e-12f);
    const float xn  = x / nrm;
    __syncthreads();

    red[tid] = xn * W[tid];                  // class 0
    __syncthreads();
    for (int off = 32; off > 0; off >>= 1) {
        if (tid < off) red[tid] += red[tid + off];
        __syncthreads();
    }
    if (tid == 0) out[b * 2 + 0] = red[0] + bias[0];
    __syncthreads();

    red[tid] = xn * W[Rq + tid];             // class 1
    __syncthreads();
    for (int off = 32; off > 0; off >>= 1) {
        if (tid < off) red[tid] += red[tid + off];
        __syncthreads();
    }
    if (tid == 0) out[b * 2 + 1] = red[0] + bias[1];
}

// ---------------------------------------------------------------------------
extern "C" void kernel_launch(void* const* d_in, const int* in_sizes, int n_in,
                              void* d_out, int out_size, void* d_ws, size_t ws_size,
                              hipStream_t stream)
{
    const int*   ids  = (const int*)  d_in[0];  // (64, 512) int32
    // d_in[1] mask, d_in[2] segment_ids, d_in[3] seq_lengths: unused
    const float* G    = (const float*)d_in[4];  // (30522, 4096) f32
    const float* G1   = (const float*)d_in[5];  // (64, 1) f32
    const float* W    = (const float*)d_in[6];  // (2, 64) f32
    const float* bias = (const float*)d_in[7];  // (2,) f32
    float*       out  = (float*)d_out;          // (64, 2) f32
    float*       ws   = (float*)d_ws;

    // Each segment column needs B * 64*64*4 = 1 MB of scratch; cap at 32 segs.
    const size_t perSegAll = (size_t)Bq * Rq * Rq * sizeof(float);
    int nseg = 32;
    while (nseg > 1 && (size_t)nseg * perSegAll > ws_size) nseg >>= 1;
    const int L = Sq / nseg;

    dim3 grid1(nseg, Bq);
    seg_matchain<<<grid1, 512, 0, stream>>>(ids, G, ws, L, nseg);
    combine_norm<<<Bq, 64, 0, stream>>>(ws, G1, W, bias, out, nseg);
}